// Aggregator1_35519379538460
// MI455X (gfx1250) — compile-verified
//
#include <hip/hip_runtime.h>

typedef float v2f __attribute__((ext_vector_type(2)));
typedef float v4f __attribute__((ext_vector_type(4)));
typedef float v8f __attribute__((ext_vector_type(8)));

#define K_NEIGH 10
#define D_EMB   128
#define L_LAB   100
#define H_HID   512
#define BB      64                        // b-rows per workgroup
#define ROWS_PER_WG (BB * K_NEIGH)        // 640 edge rows
#define TILES_PER_WG (ROWS_PER_WG / 16)   // 40 WMMA row-tiles
#define TILES_PER_WAVE (TILES_PER_WG / 8) // 5 row-tiles per wave

__device__ __forceinline__ float fast_sigmoid(float x) {
    // sigmoid(x) = 1 / (1 + exp2(-x * log2(e)))  -> v_exp_f32 + v_rcp_f32
    float e = __builtin_amdgcn_exp2f(-x * 1.4426950408889634f);
    return __builtin_amdgcn_rcpf(1.0f + e);
}

__global__ __launch_bounds__(256)
void aggregator_kernel(const int* __restrict__ nodes,
                       const int* __restrict__ neighs,
                       const float* __restrict__ embed,
                       const int* __restrict__ labels,
                       const float* __restrict__ distance,
                       const float* __restrict__ spectral,
                       const float* __restrict__ freq,
                       const float* __restrict__ dist_e,
                       const float* __restrict__ W1,
                       const float* __restrict__ b1,
                       const float* __restrict__ W2,
                       const float* __restrict__ b2,
                       float* __restrict__ out, int B)
{
    __shared__ float w_lds[ROWS_PER_WG];

    const int tid  = threadIdx.x;
    const int lane = tid & 31;
    const int wave = tid >> 5;
    const int m16  = lane & 15;
    const bool hi  = lane >= 16;

    const long wg_row0    = (long)blockIdx.x * ROWS_PER_WG;
    const long total_rows = (long)B * K_NEIGH;
    const float b2v = b2[0];

    // ---------------- Phase 1: edge MLP -> w  (V_WMMA_F32_16X16X4_F32) ----
    // Wave owns row-tiles {wave + 8*j : j=0..4}. Preload all 5 A operands,
    // then sweep the 32 hidden tiles once so W1/b1/W2 are loaded once per
    // hidden tile and reused by 5 WMMAs sharing one B operand.
    v2f a[TILES_PER_WAVE];
    #pragma unroll
    for (int j = 0; j < TILES_PER_WAVE; ++j) {
        int  t = wave + j * 8;
        long r = wg_row0 + t * 16 + m16;          // edge row handled by lane pair
        if (r >= total_rows) r = total_rows - 1;  // defensive clamp

        // A-matrix operand per ISA layout: lanes 0-15 hold features {0,1},
        // lanes 16-31 hold features {2,3} of the same 16 rows.
        if (!hi) {
            a[j].x = freq[r];                     // feature 0
            a[j].y = dist_e[r];                   // feature 1
        } else {
            int bI = (int)(r / K_NEIGH);
            int nj = neighs[r];
            int li = labels[nodes[bI]];
            int lj = labels[nj];
            a[j].x = distance[li * L_LAB + lj];   // feature 2
            a[j].y = spectral[nj];                // feature 3
        }
    }

    float acc[TILES_PER_WAVE][8];
    #pragma unroll
    for (int j = 0; j < TILES_PER_WAVE; ++j)
        #pragma unroll
        for (int v = 0; v < 8; ++v) acc[j][v] = 0.0f;

    for (int ht = 0; ht < H_HID / 16; ++ht) {
        int col = ht * 16 + m16;                  // hidden unit index for this lane
        // B operand: B[f][n] = W1[n][f]; lanes 0-15 want f={0,1}, 16-31 f={2,3}
        v2f bm = *reinterpret_cast<const v2f*>(W1 + col * 4 + (hi ? 2 : 0));
        float b1v = b1[col];
        float w2v = W2[col];

        // Fold the layer-1 bias into the WMMA C operand: C[m][n] = b1[n].
        // In the C/D layout every VGPR of a lane holds the same column n=col,
        // so a splat of b1v is exactly the bias matrix. SRC2 != VDST, so the
        // splat is built once and reused by all 5 WMMAs.
        v8f cbias;
        #pragma unroll
        for (int v = 0; v < 8; ++v) cbias[v] = b1v;

        #pragma unroll
        for (int j = 0; j < TILES_PER_WAVE; ++j) {
            v8f c = __builtin_amdgcn_wmma_f32_16x16x4_f32(false, a[j], false, bm,
                                                          (short)0, cbias,
                                                          false, false);
            // D layout: lane holds N=col, VGPR v holds row M = v (+8 for hi lanes)
            #pragma unroll
            for (int v = 0; v < 8; ++v)
                acc[j][v] += fast_sigmoid(c[v]) * w2v;
        }
    }

    // Layer 2: reduce the dot across the 16 lanes sharing each output row half
    #pragma unroll
    for (int j = 0; j < TILES_PER_WAVE; ++j) {
        #pragma unroll
        for (int v = 0; v < 8; ++v) {
            float s = acc[j][v];
            s += __shfl_xor(s, 1, 32);
            s += __shfl_xor(s, 2, 32);
            s += __shfl_xor(s, 4, 32);
            s += __shfl_xor(s, 8, 32);
            acc[j][v] = fast_sigmoid(s + b2v) * (1.0f / K_NEIGH);  // w = imp / K
        }
        if (m16 == 0) {
            int base = (wave + j * 8) * 16 + (hi ? 8 : 0);
            #pragma unroll
            for (int v = 0; v < 8; ++v) w_lds[base + v] = acc[j][v];
        }
    }

    __syncthreads();

    // ---------------- Phase 2: weighted gather-sum over embeddings --------
    // One wave per output row: 32 lanes x float4 = full 512B embed row.
    for (int i = 0; i < 8; ++i) {
        int  bl    = wave * 8 + i;                 // 0..63 local b
        long bglob = (long)blockIdx.x * BB + bl;
        if (bglob >= B) break;
        v4f vacc = {0.f, 0.f, 0.f, 0.f};
        #pragma unroll
        for (int k = 0; k < K_NEIGH; ++k) {
            int   nj = neighs[bglob * K_NEIGH + k];
            float wv = w_lds[bl * K_NEIGH + k];
            v4f e = *reinterpret_cast<const v4f*>(embed + (long)nj * D_EMB + lane * 4);
            vacc += wv * e;
        }
        *reinterpret_cast<v4f*>(out + bglob * D_EMB + lane * 4) = vacc;
    }
}

extern "C" void kernel_launch(void* const* d_in, const int* in_sizes, int n_in,
                              void* d_out, int out_size, void* d_ws, size_t ws_size,
                              hipStream_t stream) {
    const int*   nodes    = (const int*)  d_in[0];
    const int*   neighs   = (const int*)  d_in[1];
    const float* embed    = (const float*)d_in[2];
    const int*   labels   = (const int*)  d_in[3];
    const float* distance = (const float*)d_in[4];
    const float* spectral = (const float*)d_in[5];
    const float* freq     = (const float*)d_in[6];
    const float* dist_e   = (const float*)d_in[7];
    const float* W1       = (const float*)d_in[8];
    const float* b1       = (const float*)d_in[9];
    const float* W2       = (const float*)d_in[10];
    const float* b2       = (const float*)d_in[11];
    int B = in_sizes[0];

    int blocks = (B + BB - 1) / BB;   // 512 workgroups for B=32768
    aggregator_kernel<<<blocks, 256, 0, stream>>>(nodes, neighs, embed, labels,
                                                  distance, spectral, freq, dist_e,
                                                  W1, b1, W2, b2, (float*)d_out, B);
}